// m2_54546084659450
// MI455X (gfx1250) — compile-verified
//
#include <hip/hip_runtime.h>
#include <math.h>

typedef __attribute__((ext_vector_type(2))) float v2f;
typedef __attribute__((ext_vector_type(8))) float v8f;

static inline int divup_h(long a, long b) { return (int)((a + b - 1) / b); }

// ---------------- device helpers ----------------
__device__ inline float leaky02(float x) { return x > 0.f ? x : 0.2f * x; }

__device__ inline void atomicMaxF(float* addr, float val) {
  if (val >= 0.f) atomicMax((int*)addr, __float_as_int(val));
  else            atomicMin((unsigned int*)addr, (unsigned int)__float_as_int(val));
}

__device__ inline unsigned sortkey(float f) {
  unsigned u = __float_as_uint(f);
  return (u & 0x80000000u) ? ~u : (u | 0x80000000u);
}

// ---------------- generic fills ----------------
__global__ void fill_f(float* p, long n, float v) {
  long i = blockIdx.x * (long)blockDim.x + threadIdx.x;
  if (i < n) p[i] = v;
}
__global__ void fill_i(int* p, long n, int v) {
  long i = blockIdx.x * (long)blockDim.x + threadIdx.x;
  if (i < n) p[i] = v;
}

// transpose + zero-pad weights: Wt[n*F + f] = (n < K) ? W[f*K + n] : 0, n in [0,Kp)
__global__ void transpose_pad(const float* __restrict__ W, float* __restrict__ Wt,
                              int F, int K, int Kp) {
  long i = blockIdx.x * (long)blockDim.x + threadIdx.x;
  if (i >= (long)Kp * F) return;
  int n = (int)(i / F), f = (int)(i % F);
  Wt[i] = (n < K) ? W[(size_t)f * K + n] : 0.f;
}

// ---------------- WMMA fp32 GEMM: C[M,K] = A[M,F] @ W[F,K] ----------------
// Wt is the transposed, zero-padded weight [Kp,F] (Kp = roundup(K,64)).
// One wave per 16x16 tile, grid = (M/16, Kp/64), 4 waves/block.
// k-loop fully unrolled (F is a template constant); no predication in the loop:
// both A and B fragments are contiguous 8-byte pairs -> global_load_b64 + immediate offsets.
template <int F>
__global__ void gemm_wmma_f32(const float* __restrict__ A, const float* __restrict__ Wt,
                              const float* __restrict__ bias, float* __restrict__ C,
                              int K, int accumulate, int relu) {
  int wave = threadIdx.x >> 5;
  int lane = threadIdx.x & 31;
  int m0 = blockIdx.x * 16;
  int n0 = (blockIdx.y * 4 + wave) * 16;
  int c  = lane & 15;
  int hh = lane >> 4;
  int row = m0 + c;
  int col = n0 + c;                 // col < Kp always (grid covers Kp exactly)
  bool colok = col < K;

  const float* pa = A  + (size_t)row * F + 2 * hh;
  const float* pb = Wt + (size_t)col * F + 2 * hh;

  v8f acc;
  if (accumulate) {
#pragma unroll
    for (int r = 0; r < 8; ++r)
      acc[r] = colok ? C[(size_t)(m0 + r + 8 * hh) * K + col] : 0.f;
  } else {
    float b = (bias && colok) ? bias[col] : 0.f;
#pragma unroll
    for (int r = 0; r < 8; ++r) acc[r] = b;
  }

#pragma unroll
  for (int kk = 0; kk < F; kk += 4) {
    v2f a = *(const v2f*)(pa + kk);  // A rows: lanes 0-15 K{0,1}, lanes 16-31 K{2,3}
    v2f b = *(const v2f*)(pb + kk);  // B cols, same K split
    acc = __builtin_amdgcn_wmma_f32_16x16x4_f32(false, a, false, b, (short)0, acc, false, false);
  }

  if (colok) {
#pragma unroll
    for (int r = 0; r < 8; ++r) {
      float v = acc[r];
      if (relu) v = v > 0.f ? v : 0.f;
      C[(size_t)(m0 + r + 8 * hh) * K + col] = v;
    }
  }
}

// ---------------- SAGE mean aggregation ----------------
__global__ void sage_scatter(const float* __restrict__ x, const int* __restrict__ src,
                             const int* __restrict__ dst, const float* __restrict__ em,
                             float* __restrict__ s, float* __restrict__ cnt, int E, int F) {
  long g = blockIdx.x * (long)blockDim.x + threadIdx.x;
  int e = (int)(g >> 5), lane = (int)(g & 31);
  if (e >= E) return;
  float w = em ? em[e] : 1.f;
  if (w == 0.f) return;
  int sn = src[e], dn = dst[e];
  for (int f = lane; f < F; f += 32)
    atomicAdd(&s[(size_t)dn * F + f], w * x[(size_t)sn * F + f]);
  if (lane == 0) atomicAdd(&cnt[dn], w);
}

__global__ void sage_mean(float* s, const float* cnt, long total, int F) {
  long i = blockIdx.x * (long)blockDim.x + threadIdx.x;
  if (i >= total) return;
  float c = cnt[i / F];
  c = c > 1.f ? c : 1.f;
  s[i] = s[i] / c;
}

__global__ void relu_add(const float* t1, const float* t2, float* out, long n) {
  long i = blockIdx.x * (long)blockDim.x + threadIdx.x;
  if (i >= n) return;
  float a = t1[i];
  out[i] = (a > 0.f ? a : 0.f) + t2[i];
}

// ---------------- TopK pooling ----------------
__global__ void vec_norm(const float* w, int n, float* out) {
  __shared__ float sh[256];
  float a = 0.f;
  for (int i = threadIdx.x; i < n; i += blockDim.x) a += w[i] * w[i];
  sh[threadIdx.x] = a;
  __syncthreads();
  for (int s = 128; s > 0; s >>= 1) {
    if ((int)threadIdx.x < s) sh[threadIdx.x] += sh[threadIdx.x + s];
    __syncthreads();
  }
  if (threadIdx.x == 0) *out = sqrtf(sh[0]);
}

__global__ void pool_score(const float* h, const float* w, const float* normp,
                           float* score, unsigned* keys, int n, int F) {
  int i = blockIdx.x * blockDim.x + threadIdx.x;
  if (i >= n) return;
  float d = 0.f;
  for (int f = 0; f < F; ++f) d += h[(size_t)i * F + f] * w[f];
  float s = tanhf(d / *normp);
  score[i] = s;
  keys[i] = sortkey(s);
}

// exact k-th largest key via 4x8-bit MSB-first radix select (single block of 256)
__global__ void radix_kth(const unsigned* keys, int n, int k, unsigned* kth_out) {
  __shared__ unsigned hist[256];
  __shared__ unsigned prefix;
  __shared__ int kk;
  if (threadIdx.x == 0) { prefix = 0u; kk = k; }
  __syncthreads();
  for (int pass = 0; pass < 4; ++pass) {
    int shift = 24 - 8 * pass;
    hist[threadIdx.x] = 0u;
    __syncthreads();
    unsigned pref = prefix;
    for (int i = threadIdx.x; i < n; i += blockDim.x) {
      unsigned key = keys[i];
      bool match = (pass == 0) || ((key >> (shift + 8)) == (pref >> (shift + 8)));
      if (match) atomicAdd(&hist[(key >> shift) & 255u], 1u);
    }
    __syncthreads();
    if (threadIdx.x == 0) {
      int rem = kk;
      for (int b = 255; b >= 0; --b) {
        int hc = (int)hist[b];
        if (rem <= hc) { prefix |= ((unsigned)b) << shift; break; }
        rem -= hc;
      }
      kk = rem;
    }
    __syncthreads();
  }
  if (threadIdx.x == 0) *kth_out = prefix;
}

__global__ void count_gt(const unsigned* keys, int n, const unsigned* kth, int* cntG) {
  int i = blockIdx.x * blockDim.x + threadIdx.x;
  if (i < n && keys[i] > *kth) atomicAdd(cntG, 1);
}

__global__ void pool_compact(const float* h, const float* score, const unsigned* keys,
                             const unsigned* kth, const int* cntG, int* cc,
                             int* newid, float* keep, float* xout, int n, int k, int F) {
  int i = blockIdx.x * blockDim.x + threadIdx.x;
  if (i >= n) return;
  unsigned key = keys[i], kv = *kth;
  int pos = -1;
  if (key > kv) pos = atomicAdd(&cc[0], 1);
  else if (key == kv) {
    int t = atomicAdd(&cc[1], 1);
    int p = *cntG + t;
    if (p < k) pos = p;
  }
  if (pos >= 0) {
    newid[i] = pos;
    keep[i] = 1.f;
    float sv = score[i];
    for (int f = 0; f < F; ++f) xout[(size_t)pos * F + f] = h[(size_t)i * F + f] * sv;
  } else {
    newid[i] = 0;
    keep[i] = 0.f;
  }
}

__global__ void edge_remap(const int* src, const int* dst, const float* em,
                           const int* newid, const float* keep,
                           int* src2, int* dst2, float* em2, int E) {
  int e = blockIdx.x * blockDim.x + threadIdx.x;
  if (e >= E) return;
  int s = src[e], d = dst[e];
  em2[e] = (em ? em[e] : 1.f) * keep[s] * keep[d];
  src2[e] = newid[s];
  dst2[e] = newid[d];
}

// ---------------- GAT ----------------
#define HEADS 8
#define H2C 128

__global__ void gat_alad(const float* __restrict__ xh, const float* __restrict__ asrc,
                         const float* __restrict__ adst, float* al, float* ad, int n) {
  int i = blockIdx.x * blockDim.x + threadIdx.x;
  if (i >= n * HEADS) return;
  int node = i >> 3, h = i & 7;
  const float* row = xh + (size_t)node * (HEADS * H2C) + h * H2C;
  float a = 0.f, b = 0.f;
  for (int c = 0; c < H2C; ++c) {
    float v = row[c];
    a += v * asrc[h * H2C + c];
    b += v * adst[h * H2C + c];
  }
  al[i] = a;
  ad[i] = b;
}

__global__ void gat_mx_init(const float* al, const float* ad, float* mx, int n) {
  int i = blockIdx.x * blockDim.x + threadIdx.x;
  if (i < n * HEADS) mx[i] = leaky02(al[i] + ad[i]);   // self loop
}

__global__ void gat_edge_max(const int* src, const int* dst, const float* em,
                             const float* al, const float* ad, float* mx, int E) {
  int i = blockIdx.x * blockDim.x + threadIdx.x;
  if (i >= E * HEADS) return;
  int e = i >> 3, h = i & 7;
  if (em[e] == 0.f) return;
  float v = leaky02(al[src[e] * HEADS + h] + ad[dst[e] * HEADS + h]);
  atomicMaxF(&mx[dst[e] * HEADS + h], v);
}

__global__ void gat_ssum_init(const float* al, const float* ad, const float* mx,
                              float* ssum, int n) {
  int i = blockIdx.x * blockDim.x + threadIdx.x;
  if (i < n * HEADS) ssum[i] = expf(leaky02(al[i] + ad[i]) - mx[i]);
}

__global__ void gat_edge_sum(const int* src, const int* dst, const float* em,
                             const float* al, const float* ad, const float* mx,
                             float* ssum, int E) {
  int i = blockIdx.x * blockDim.x + threadIdx.x;
  if (i >= E * HEADS) return;
  int e = i >> 3, h = i & 7;
  if (em[e] == 0.f) return;
  int dn = dst[e];
  float v = leaky02(al[src[e] * HEADS + h] + ad[dn * HEADS + h]);
  atomicAdd(&ssum[dn * HEADS + h], expf(v - mx[dn * HEADS + h]));
}

// self-loop contribution; writes (initializes) gato fully
__global__ void gat_self_agg(const float* xh, const float* al, const float* ad,
                             const float* mx, const float* ssum, float* gato, int n) {
  long i = blockIdx.x * (long)blockDim.x + threadIdx.x;
  if (i >= (long)n * H2C) return;
  int node = (int)(i >> 7), c = (int)(i & 127);
  float acc = 0.f;
  for (int h = 0; h < HEADS; ++h) {
    float selfv = leaky02(al[node * HEADS + h] + ad[node * HEADS + h]);
    float w = expf(selfv - mx[node * HEADS + h]) / ssum[node * HEADS + h];
    acc += w * xh[(size_t)node * (HEADS * H2C) + h * H2C + c];
  }
  gato[i] = acc;
}

// 32-thread group per (edge, head); 4 channels per lane
__global__ void gat_edge_agg(const int* src, const int* dst, const float* em,
                             const float* al, const float* ad, const float* mx,
                             const float* ssum, const float* xh, float* gato, int E) {
  long g = blockIdx.x * (long)blockDim.x + threadIdx.x;
  long grp = g >> 5;
  int lane = (int)(g & 31);
  if (grp >= (long)E * HEADS) return;
  int e = (int)(grp >> 3), h = (int)(grp & 7);
  if (em[e] == 0.f) return;
  int sn = src[e], dn = dst[e];
  float v = leaky02(al[sn * HEADS + h] + ad[dn * HEADS + h]);
  float w = expf(v - mx[dn * HEADS + h]) / ssum[dn * HEADS + h];
#pragma unroll
  for (int j = 0; j < 4; ++j) {
    int c = lane + 32 * j;
    atomicAdd(&gato[(size_t)dn * H2C + c], w * xh[(size_t)sn * (HEADS * H2C) + h * H2C + c]);
  }
}

// relu(mean_heads + bias) + residual
__global__ void gat_finalize(const float* gato, const float* b, const float* resid,
                             float* out, long n) {
  long i = blockIdx.x * (long)blockDim.x + threadIdx.x;
  if (i >= n) return;
  int c = (int)(i & 127);
  float v = gato[i] * 0.125f + b[c];
  v = v > 0.f ? v : 0.f;
  out[i] = v + resid[i];
}

// ---------------- GCN ----------------
__global__ void deg_add(const int* dst, const float* em, float* deg, int E) {
  int e = blockIdx.x * blockDim.x + threadIdx.x;
  if (e >= E) return;
  float w = em[e];
  if (w != 0.f) atomicAdd(&deg[dst[e]], w);
}

__global__ void rsqrt_from(const float* deg, float* dinv, int n) {
  int i = blockIdx.x * blockDim.x + threadIdx.x;
  if (i < n) dinv[i] = rsqrtf(deg[i]);
}

__global__ void gcn_self(const float* xw, const float* dinv, float* out, int n, int F) {
  long i = blockIdx.x * (long)blockDim.x + threadIdx.x;
  if (i >= (long)n * F) return;
  float di = dinv[i / F];
  out[i] = di * di * xw[i];
}

__global__ void gcn_edge(const int* src, const int* dst, const float* em,
                         const float* dinv, const float* xw, float* out, int E, int F) {
  long g = blockIdx.x * (long)blockDim.x + threadIdx.x;
  int e = (int)(g >> 5), lane = (int)(g & 31);
  if (e >= E) return;
  float w = em[e];
  if (w == 0.f) return;
  int sn = src[e], dn = dst[e];
  float coef = w * dinv[sn] * dinv[dn];
  for (int f = lane; f < F; f += 32)
    atomicAdd(&out[(size_t)dn * F + f], coef * xw[(size_t)sn * F + f]);
}

__global__ void bias_act(float* p, const float* b, long n, int F, int relu) {
  long i = blockIdx.x * (long)blockDim.x + threadIdx.x;
  if (i >= n) return;
  float v = p[i] + b[i % F];
  if (relu) v = v > 0.f ? v : 0.f;
  p[i] = v;
}

// ---------------- final reduce + log_softmax ----------------
#define NCLS 40
__global__ void colsum40(const float* in, int n, float* g) {
  __shared__ float sh[NCLS];
  if (threadIdx.x < NCLS) sh[threadIdx.x] = 0.f;
  __syncthreads();
  int i = blockIdx.x * blockDim.x + threadIdx.x;
  if (i < n)
    for (int c = 0; c < NCLS; ++c) atomicAdd(&sh[c], in[(size_t)i * NCLS + c]);
  __syncthreads();
  if (threadIdx.x < NCLS) atomicAdd(&g[threadIdx.x], sh[threadIdx.x]);
}

__global__ void logsoftmax40(const float* g, float* out, float invK) {
  if (threadIdx.x == 0 && blockIdx.x == 0) {
    float v[NCLS], m = -1e30f;
    for (int c = 0; c < NCLS; ++c) { v[c] = g[c] * invK; if (v[c] > m) m = v[c]; }
    float s = 0.f;
    for (int c = 0; c < NCLS; ++c) s += expf(v[c] - m);
    float ls = logf(s);
    for (int c = 0; c < NCLS; ++c) out[c] = v[c] - m - ls;
  }
}

// ---------------- host launch ----------------
extern "C" void kernel_launch(void* const* d_in, const int* in_sizes, int n_in,
                              void* d_out, int out_size, void* d_ws, size_t ws_size,
                              hipStream_t stream) {
  (void)in_sizes; (void)n_in; (void)out_size; (void)ws_size;
  const int N = 20000, E = 320000, F_IN = 128, H1 = 64, H2 = 128;
  const int K1 = 16000, K2 = 12800, NC = 40;
  const int TB = 256;

  const float* x        = (const float*)d_in[0];
  const int*   ei       = (const int*)d_in[1];
  const int*   src      = ei;
  const int*   dst      = ei + E;
  const float* sage1_wl = (const float*)d_in[3];
  const float* sage1_bl = (const float*)d_in[4];
  const float* sage1_wr = (const float*)d_in[5];
  const float* res1_w   = (const float*)d_in[6];
  const float* res1_b   = (const float*)d_in[7];
  const float* pool1_w  = (const float*)d_in[8];
  const float* sage2_wl = (const float*)d_in[9];
  const float* sage2_bl = (const float*)d_in[10];
  const float* sage2_wr = (const float*)d_in[11];
  const float* res2_w   = (const float*)d_in[12];
  const float* res2_b   = (const float*)d_in[13];
  const float* pool2_w  = (const float*)d_in[14];
  const float* gat_w    = (const float*)d_in[15];
  const float* gat_asrc = (const float*)d_in[16];
  const float* gat_adst = (const float*)d_in[17];
  const float* gat_b    = (const float*)d_in[18];
  const float* gcn1_w   = (const float*)d_in[19];
  const float* gcn1_b   = (const float*)d_in[20];
  const float* gcn2_w   = (const float*)d_in[21];
  const float* gcn2_b   = (const float*)d_in[22];

  char* base = (char*)d_ws;
  size_t off = 0;
  auto alloc = [&](size_t bytes) -> void* {
    void* p = base + off;
    off += (bytes + 255) & ~(size_t)255;
    return p;
  };

  float*    s1     = (float*)alloc((size_t)N * F_IN * 4);
  float*    c1     = (float*)alloc((size_t)N * 4);
  float*    t1     = (float*)alloc((size_t)N * H1 * 4);
  float*    t2     = (float*)alloc((size_t)N * H1 * 4);
  float*    h1     = (float*)alloc((size_t)N * H1 * 4);
  float*    score1 = (float*)alloc((size_t)N * 4);
  unsigned* keys1  = (unsigned*)alloc((size_t)N * 4);
  int*      newid1 = (int*)alloc((size_t)N * 4);
  float*    keep1  = (float*)alloc((size_t)N * 4);
  float*    h1p    = (float*)alloc((size_t)K1 * H1 * 4);
  int*      srcB   = (int*)alloc((size_t)E * 4);
  int*      dstB   = (int*)alloc((size_t)E * 4);
  float*    emB    = (float*)alloc((size_t)E * 4);
  float*    s2     = (float*)alloc((size_t)K1 * H1 * 4);
  float*    c2     = (float*)alloc((size_t)K1 * 4);
  float*    t1b    = (float*)alloc((size_t)K1 * H2 * 4);
  float*    t2b    = (float*)alloc((size_t)K1 * H2 * 4);
  float*    h2     = (float*)alloc((size_t)K1 * H2 * 4);
  float*    score2 = (float*)alloc((size_t)K1 * 4);
  unsigned* keys2  = (unsigned*)alloc((size_t)K1 * 4);
  int*      newid2 = (int*)alloc((size_t)K1 * 4);
  float*    keep2  = (float*)alloc((size_t)K1 * 4);
  float*    h2p    = (float*)alloc((size_t)K2 * H2 * 4);
  int*      srcC   = (int*)alloc((size_t)E * 4);
  int*      dstC   = (int*)alloc((size_t)E * 4);
  float*    emC    = (float*)alloc((size_t)E * 4);
  float*    xh     = (float*)alloc((size_t)K2 * HEADS * H2 * 4);
  float*    al     = (float*)alloc((size_t)K2 * HEADS * 4);
  float*    ad     = (float*)alloc((size_t)K2 * HEADS * 4);
  float*    mx     = (float*)alloc((size_t)K2 * HEADS * 4);
  float*    ssum   = (float*)alloc((size_t)K2 * HEADS * 4);
  float*    gato   = (float*)alloc((size_t)K2 * H2 * 4);
  float*    g1in   = (float*)alloc((size_t)K2 * H2 * 4);
  float*    deg    = (float*)alloc((size_t)K2 * 4);
  float*    dinv   = (float*)alloc((size_t)K2 * 4);
  float*    xw     = (float*)alloc((size_t)K2 * H1 * 4);
  float*    gc1    = (float*)alloc((size_t)K2 * H1 * 4);
  float*    xw2    = (float*)alloc((size_t)K2 * NC * 4);
  float*    gc2    = (float*)alloc((size_t)K2 * NC * 4);
  float*    wt     = (float*)alloc((size_t)1024 * 128 * 4);   // shared transposed-weight scratch
  float*    gvec   = (float*)alloc((size_t)NC * 4);
  float*    normp  = (float*)alloc(4);
  unsigned* kth    = (unsigned*)alloc(4);
  int*      cnts   = (int*)alloc(3 * 4);   // [0]=cntGreater, [1..2]=compaction counters

  // GEMM helper: Kp = roundup(K,64); transpose+pad weights, then WMMA GEMM.
  auto gemmF128 = [&](const float* A, const float* W, const float* bias, float* C,
                      int M, int K, int accum, int relu) {
    int Kp = ((K + 63) / 64) * 64;
    transpose_pad<<<divup_h((long)Kp * 128, TB), TB, 0, stream>>>(W, wt, 128, K, Kp);
    gemm_wmma_f32<128><<<dim3(M / 16, Kp / 64), 128, 0, stream>>>(A, wt, bias, C, K, accum, relu);
  };
  auto gemmF64 = [&](const float* A, const float* W, const float* bias, float* C,
                     int M, int K, int accum, int relu) {
    int Kp = ((K + 63) / 64) * 64;
    transpose_pad<<<divup_h((long)Kp * 64, TB), TB, 0, stream>>>(W, wt, 64, K, Kp);
    gemm_wmma_f32<64><<<dim3(M / 16, Kp / 64), 128, 0, stream>>>(A, wt, bias, C, K, accum, relu);
  };

  // ---- stage 1: SAGE1 on (N, F_IN) -> h1[N,H1] ----
  fill_f<<<divup_h((long)N * F_IN, TB), TB, 0, stream>>>(s1, (long)N * F_IN, 0.f);
  fill_f<<<divup_h(N, TB), TB, 0, stream>>>(c1, N, 0.f);
  sage_scatter<<<divup_h((long)E * 32, TB), TB, 0, stream>>>(x, src, dst, nullptr, s1, c1, E, F_IN);
  sage_mean<<<divup_h((long)N * F_IN, TB), TB, 0, stream>>>(s1, c1, (long)N * F_IN, F_IN);
  gemmF128(s1, sage1_wl, sage1_bl, t1, N, H1, 0, 0);
  gemmF128(x, sage1_wr, nullptr, t1, N, H1, 1, 0);
  gemmF128(x, res1_w, res1_b, t2, N, H1, 0, 0);
  relu_add<<<divup_h((long)N * H1, TB), TB, 0, stream>>>(t1, t2, h1, (long)N * H1);

  // ---- pool1: keep K1 of N ----
  vec_norm<<<1, 256, 0, stream>>>(pool1_w, H1, normp);
  pool_score<<<divup_h(N, TB), TB, 0, stream>>>(h1, pool1_w, normp, score1, keys1, N, H1);
  radix_kth<<<1, 256, 0, stream>>>(keys1, N, K1, kth);
  fill_i<<<1, 64, 0, stream>>>(cnts, 3, 0);
  count_gt<<<divup_h(N, TB), TB, 0, stream>>>(keys1, N, kth, cnts);
  pool_compact<<<divup_h(N, TB), TB, 0, stream>>>(h1, score1, keys1, kth, cnts, cnts + 1,
                                                  newid1, keep1, h1p, N, K1, H1);
  edge_remap<<<divup_h(E, TB), TB, 0, stream>>>(src, dst, nullptr, newid1, keep1, srcB, dstB, emB, E);

  // ---- stage 2: SAGE2 on (K1, H1) -> h2[K1,H2] ----
  fill_f<<<divup_h((long)K1 * H1, TB), TB, 0, stream>>>(s2, (long)K1 * H1, 0.f);
  fill_f<<<divup_h(K1, TB), TB, 0, stream>>>(c2, K1, 0.f);
  sage_scatter<<<divup_h((long)E * 32, TB), TB, 0, stream>>>(h1p, srcB, dstB, emB, s2, c2, E, H1);
  sage_mean<<<divup_h((long)K1 * H1, TB), TB, 0, stream>>>(s2, c2, (long)K1 * H1, H1);
  gemmF64(s2, sage2_wl, sage2_bl, t1b, K1, H2, 0, 0);
  gemmF64(h1p, sage2_wr, nullptr, t1b, K1, H2, 1, 0);
  gemmF64(h1p, res2_w, res2_b, t2b, K1, H2, 0, 0);
  relu_add<<<divup_h((long)K1 * H2, TB), TB, 0, stream>>>(t1b, t2b, h2, (long)K1 * H2);

  // ---- pool2: keep K2 of K1 ----
  vec_norm<<<1, 256, 0, stream>>>(pool2_w, H2, normp);
  pool_score<<<divup_h(K1, TB), TB, 0, stream>>>(h2, pool2_w, normp, score2, keys2, K1, H2);
  radix_kth<<<1, 256, 0, stream>>>(keys2, K1, K2, kth);
  fill_i<<<1, 64, 0, stream>>>(cnts, 3, 0);
  count_gt<<<divup_h(K1, TB), TB, 0, stream>>>(keys2, K1, kth, cnts);
  pool_compact<<<divup_h(K1, TB), TB, 0, stream>>>(h2, score2, keys2, kth, cnts, cnts + 1,
                                                   newid2, keep2, h2p, K1, K2, H2);
  edge_remap<<<divup_h(E, TB), TB, 0, stream>>>(srcB, dstB, emB, newid2, keep2, srcC, dstC, emC, E);

  // ---- GAT on (K2, H2); h2p is also the residual x1 ----
  gemmF128(h2p, gat_w, nullptr, xh, K2, HEADS * H2, 0, 0);
  gat_alad<<<divup_h((long)K2 * HEADS, TB), TB, 0, stream>>>(xh, gat_asrc, gat_adst, al, ad, K2);
  gat_mx_init<<<divup_h((long)K2 * HEADS, TB), TB, 0, stream>>>(al, ad, mx, K2);
  gat_edge_max<<<divup_h((long)E * HEADS, TB), TB, 0, stream>>>(srcC, dstC, emC, al, ad, mx, E);
  gat_ssum_init<<<divup_h((long)K2 * HEADS, TB), TB, 0, stream>>>(al, ad, mx, ssum, K2);
  gat_edge_sum<<<divup_h((long)E * HEADS, TB), TB, 0, stream>>>(srcC, dstC, emC, al, ad, mx, ssum, E);
  gat_self_agg<<<divup_h((long)K2 * H2C, TB), TB, 0, stream>>>(xh, al, ad, mx, ssum, gato, K2);
  gat_edge_agg<<<divup_h((long)E * HEADS * 32, TB), TB, 0, stream>>>(srcC, dstC, emC, al, ad, mx, ssum, xh, gato, E);
  gat_finalize<<<divup_h((long)K2 * H2C, TB), TB, 0, stream>>>(gato, gat_b, h2p, g1in, (long)K2 * H2C);

  // ---- GCN degree normalization (shared by both GCN layers) ----
  fill_f<<<divup_h(K2, TB), TB, 0, stream>>>(deg, K2, 1.f);   // self loop
  deg_add<<<divup_h(E, TB), TB, 0, stream>>>(dstC, emC, deg, E);
  rsqrt_from<<<divup_h(K2, TB), TB, 0, stream>>>(deg, dinv, K2);

  // ---- GCN1: (g1in @ gcn1_w), sym-norm aggregate, +b, relu ----
  gemmF128(g1in, gcn1_w, nullptr, xw, K2, H1, 0, 0);
  gcn_self<<<divup_h((long)K2 * H1, TB), TB, 0, stream>>>(xw, dinv, gc1, K2, H1);
  gcn_edge<<<divup_h((long)E * 32, TB), TB, 0, stream>>>(srcC, dstC, emC, dinv, xw, gc1, E, H1);
  bias_act<<<divup_h((long)K2 * H1, TB), TB, 0, stream>>>(gc1, gcn1_b, (long)K2 * H1, H1, 1);

  // ---- GCN2: (gc1 @ gcn2_w), aggregate, +b ----
  gemmF64(gc1, gcn2_w, nullptr, xw2, K2, NC, 0, 0);
  gcn_self<<<divup_h((long)K2 * NC, TB), TB, 0, stream>>>(xw2, dinv, gc2, K2, NC);
  gcn_edge<<<divup_h((long)E * 32, TB), TB, 0, stream>>>(srcC, dstC, emC, dinv, xw2, gc2, E, NC);
  bias_act<<<divup_h((long)K2 * NC, TB), TB, 0, stream>>>(gc2, gcn2_b, (long)K2 * NC, NC, 0);

  // ---- global mean pool + log_softmax ----
  fill_f<<<1, 64, 0, stream>>>(gvec, NC, 0.f);
  colsum40<<<divup_h(K2, TB), TB, 0, stream>>>(gc2, K2, gvec);
  logsoftmax40<<<1, 64, 0, stream>>>(gvec, (float*)d_out, 1.f / (float)K2);
}